// PlaceCells_1503238553823
// MI455X (gfx1250) — compile-verified
//
#include <hip/hip_runtime.h>
#include <stdint.h>

#define CELL_DIM 64
#define NUM_CELLS 512
#define CSLICES 4
#define CELLS_PER_SLICE (NUM_CELLS / CSLICES) /* 128 cells -> 32KB LDS slice */

typedef float v2f __attribute__((ext_vector_type(2)));
typedef float v8f __attribute__((ext_vector_type(8)));
typedef unsigned int v4u __attribute__((ext_vector_type(4)));
typedef int v8i __attribute__((ext_vector_type(8)));
typedef int v4i __attribute__((ext_vector_type(4)));

// ---------------------------------------------------------------------------
// Kernel 1: scores[n,c] = -0.5 * (sum_k |x[n,k] - p[c,k]|)^2
// One thread per row, 4 cell-slices across blockIdx.y (fills more WGPs).
// placeCells slice staged to LDS via the Tensor Data Mover; p reads are
// wave-uniform LDS broadcasts (all lanes iterate the same (c,k)).
// ---------------------------------------------------------------------------
__global__ __launch_bounds__(256) void pc_scores_kernel(
    const float* __restrict__ x, const float* __restrict__ cells,
    float* __restrict__ out, int rows)
{
    __shared__ float lds_p[CELLS_PER_SLICE * CELL_DIM]; // 32 KB
    const int j = blockIdx.y;
    const float* slice = cells + (size_t)j * CELLS_PER_SLICE * CELL_DIM;

#if __has_builtin(__builtin_amdgcn_tensor_load_to_lds)
    if (threadIdx.x < 32) { // wave 0 issues one TDM DMA for the whole 32KB tile
        unsigned lds_off = (unsigned)(unsigned long long)(uintptr_t)(void*)&lds_p[0];
        unsigned long long ga = (unsigned long long)(uintptr_t)slice;
        v4u g0;
        g0[0] = 1u;                                   // count=1, user descriptor
        g0[1] = lds_off;                              // lds_addr (bytes)
        g0[2] = (unsigned)(ga & 0xffffffffull);       // global_addr[31:0]
        g0[3] = (unsigned)((ga >> 32) & 0x1ffffffull) // global_addr[56:32]
                | 0x80000000u;                        // type=2 ("image")
        v8i g1;
        g1[0] = (int)(2u << 16);                      // data_size = 4 bytes
        g1[1] = (int)((CELL_DIM & 0xffffu) << 16);    // tensor_dim0 lo16 @ [63:48]
        g1[2] = (int)((CELLS_PER_SLICE & 0xffffu) << 16); // tensor_dim1 lo16 @ [95:80]
        g1[3] = (int)((CELL_DIM & 0xffffu) << 16);    // tile_dim0 @ [127:112]
        g1[4] = (int)(CELLS_PER_SLICE & 0xffffu);     // tile_dim1 (tile_dim2=0)
        g1[5] = (int)CELL_DIM;                        // tensor_dim0_stride lo32
        g1[6] = 0;
        g1[7] = 0;
        v4i g2 = {0, 0, 0, 0};
        v4i g3 = {0, 0, 0, 0};
        v8i g4 = {0, 0, 0, 0, 0, 0, 0, 0};            // extra group (clang-23 form)
        __builtin_amdgcn_tensor_load_to_lds(g0, g1, g2, g3, g4, 0);
        __builtin_amdgcn_s_wait_tensorcnt(0);
    }
#else
    for (int i = threadIdx.x; i < CELLS_PER_SLICE * CELL_DIM / 4; i += 256)
        ((float4*)lds_p)[i] = ((const float4*)slice)[i];
#endif
    __syncthreads();

    const int n = blockIdx.x * 256 + threadIdx.x;
    if (n >= rows) return;

    float xr[CELL_DIM]; // full row resident in VGPRs
    #pragma unroll
    for (int i = 0; i < CELL_DIM / 4; ++i)
        ((float4*)xr)[i] = ((const float4*)(x + (size_t)n * CELL_DIM))[i];

    float* orow = out + (size_t)n * NUM_CELLS + j * CELLS_PER_SLICE;
    for (int c = 0; c < CELLS_PER_SLICE; ++c) {
        const float4* p4 = (const float4*)(lds_p + c * CELL_DIM);
        float acc = 0.f;
        #pragma unroll
        for (int kk = 0; kk < CELL_DIM / 4; ++kk) {
            float4 p = p4[kk]; // ds_load_b128, wave-uniform broadcast
            acc += fabsf(xr[4 * kk + 0] - p.x);
            acc += fabsf(xr[4 * kk + 1] - p.y);
            acc += fabsf(xr[4 * kk + 2] - p.z);
            acc += fabsf(xr[4 * kk + 3] - p.w);
        }
        orow[c] = -0.5f * acc * acc;
    }
}

// ---------------------------------------------------------------------------
// Kernel 2: in-place stable softmax over 512 columns. 16 rows per block.
// Each wave owns a 16-row x 64-col tile laid out as WMMA A-fragments:
//   lane L<16 : row m=L,   holds cols 4t+{0,1}  (k=0,1)
//   lane L>=16: row m=L-16,holds cols 4t+{2,3}  (k=2,3)
// Row-sums of exp() computed by 16 chained V_WMMA_F32_16X16X4_F32 with B=1.
// ---------------------------------------------------------------------------
__global__ __launch_bounds__(256) void pc_softmax_kernel(float* __restrict__ inout,
                                                         int rows)
{
    __shared__ float pmax[16][16]; // [wave*2+half][row]
    __shared__ float rmax[16];
    __shared__ float rsum[16];

    const int tid  = threadIdx.x;
    const int w    = tid >> 5;
    const int L    = tid & 31;
    const int half = L >> 4;
    const int m    = L & 15;
    const int r0   = blockIdx.x * 16;
    (void)rows;

    float* rowp = inout + (size_t)(r0 + m) * NUM_CELLS + w * 64 + half * 2;

    float2 v[16];
    #pragma unroll
    for (int t = 0; t < 16; ++t) v[t] = *(const float2*)(rowp + 4 * t);

    if (tid < 16) rsum[tid] = 0.f;

    float mx = -3.402823466e38f;
    #pragma unroll
    for (int t = 0; t < 16; ++t) mx = fmaxf(mx, fmaxf(v[t].x, v[t].y));
    pmax[w * 2 + half][m] = mx;
    __syncthreads();

    if (tid < 16) {
        float q = -3.402823466e38f;
        #pragma unroll
        for (int i = 0; i < 16; ++i) q = fmaxf(q, pmax[i][tid]);
        rmax[tid] = q;
    }
    __syncthreads();

    const float rm = rmax[m];
    float2 e[16];
    #pragma unroll
    for (int t = 0; t < 16; ++t) {
        e[t].x = __expf(v[t].x - rm);
        e[t].y = __expf(v[t].y - rm);
    }

#if __has_builtin(__builtin_amdgcn_wmma_f32_16x16x4_f32)
    // D[m,n] = sum_k A[m,k] * 1 + C : per-row partial sums across the wave.
    v8f cacc = {0.f, 0.f, 0.f, 0.f, 0.f, 0.f, 0.f, 0.f};
    v2f bones = {1.f, 1.f};
    #pragma unroll
    for (int t = 0; t < 16; ++t) {
        v2f a = {e[t].x, e[t].y};
        cacc = __builtin_amdgcn_wmma_f32_16x16x4_f32(
            false, a, false, bones, (short)0, cacc, false, false);
    }
    // C/D layout: VGPR j -> M=j (lanes 0-15) / M=j+8 (lanes 16-31), any N.
    if (m == 0) {
        #pragma unroll
        for (int jj = 0; jj < 8; ++jj)
            atomicAdd(&rsum[half * 8 + jj], cacc[jj]); // ds_add_f32
    }
#else
    float ls = 0.f;
    #pragma unroll
    for (int t = 0; t < 16; ++t) ls += e[t].x + e[t].y;
    atomicAdd(&rsum[m], ls);
#endif
    __syncthreads();

    const float inv = 1.0f / rsum[m];
    #pragma unroll
    for (int t = 0; t < 16; ++t) {
        float2 o;
        o.x = e[t].x * inv;
        o.y = e[t].y * inv;
        *(float2*)(rowp + 4 * t) = o;
    }
}

extern "C" void kernel_launch(void* const* d_in, const int* in_sizes, int n_in,
                              void* d_out, int out_size, void* d_ws, size_t ws_size,
                              hipStream_t stream) {
    (void)n_in; (void)out_size; (void)d_ws; (void)ws_size;
    const float* x     = (const float*)d_in[0];
    const float* cells = (const float*)d_in[1];
    float* out = (float*)d_out;
    const int rows = in_sizes[0] / CELL_DIM; // 8192

    dim3 g1((rows + 255) / 256, CSLICES);
    pc_scores_kernel<<<g1, 256, 0, stream>>>(x, cells, out, rows);
    pc_softmax_kernel<<<rows / 16, 256, 0, stream>>>(out, rows);
}